// GraphWaveNet_20804821582312
// MI455X (gfx1250) — compile-verified
//
#include <hip/hip_runtime.h>

typedef __attribute__((ext_vector_type(2))) float v2f;
typedef __attribute__((ext_vector_type(8))) float v8f;

#define NB 8
#define NT 13
#define NN 1000
#define NE 16000
#define NEN 17000   // NE + NN self loops

__device__ inline v8f wmma4(v2f a, v2f b, v8f c) {
  // D(16x16,f32) = A(16x4,f32) * B(4x16,f32) + C   (wave32, fp32 precision)
  return __builtin_amdgcn_wmma_f32_16x16x4_f32(false, a, false, b, (short)0, c,
                                               false, false);
}

// ---------------- static-feature MLP: s = relu((static@w1+b1)@w2+b2) --------
__global__ __launch_bounds__(64) void k_static(const float* __restrict__ st,
                                               const float* __restrict__ w1,
                                               const float* __restrict__ b1,
                                               const float* __restrict__ w2,
                                               const float* __restrict__ b2,
                                               float* __restrict__ s) {
  __shared__ float hid[64];
  int n = blockIdx.x, t = threadIdx.x;
  float acc = b1[t];
  for (int c = 0; c < 8; ++c) acc += st[n * 8 + c] * w1[c * 64 + t];
  hid[t] = acc;
  __syncthreads();
  if (t < 32) {
    float a2 = b2[t];
    for (int c = 0; c < 64; ++c) a2 += hid[c] * w2[c * 32 + t];
    s[n * 32 + t] = fmaxf(a2, 0.f);
  }
}

// --------- input projection: x(b,n,t,r) = dyn(b,t,n,:)@w_in + b_in + s(n,r) -
__global__ void k_input(const float* __restrict__ dyn,
                        const float* __restrict__ w_in,
                        const float* __restrict__ b_in,
                        const float* __restrict__ s, float* __restrict__ x) {
  long id = (long)blockIdx.x * 256 + threadIdx.x;
  long total = (long)NB * NN * NT * 32;
  if (id >= total) return;
  int r = (int)(id & 31);
  long p = id >> 5;                 // pixel = (b*NN+n)*NT + t
  int t = (int)(p % NT);
  long q = p / NT;
  int n = (int)(q % NN), b = (int)(q / NN);
  const float* dp = dyn + (((long)b * NT + t) * NN + n) * 2;
  x[id] = dp[0] * w_in[r] + dp[1] * w_in[32 + r] + b_in[r] + s[n * 32 + r];
}

// ------------------- CSR build over (edges + self loops), dst-sorted --------
__global__ void k_edges(const int* __restrict__ ei, int* __restrict__ srcA,
                        int* __restrict__ dstA) {
  int e = blockIdx.x * 256 + threadIdx.x;
  if (e >= NEN) return;
  srcA[e] = (e < NE) ? ei[e] : (e - NE);
  dstA[e] = (e < NE) ? ei[NE + e] : (e - NE);
}
__global__ void k_deg(const int* __restrict__ dstA, int* __restrict__ deg) {
  int e = blockIdx.x * 256 + threadIdx.x;
  if (e >= NEN) return;
  atomicAdd(&deg[dstA[e]], 1);
}
__global__ __launch_bounds__(1024) void k_scan(const int* __restrict__ deg,
                                               int* __restrict__ row_start) {
  __shared__ int sc[1024];
  int t = threadIdx.x;
  sc[t] = (t < NN) ? deg[t] : 0;
  __syncthreads();
  for (int o = 1; o < 1024; o <<= 1) {
    int v = (t >= o) ? sc[t - o] : 0;
    __syncthreads();
    sc[t] += v;
    __syncthreads();
  }
  if (t == 0) row_start[0] = 0;
  if (t < NN) row_start[t + 1] = sc[t];
}
__global__ void k_fill(const int* __restrict__ srcA, const int* __restrict__ dstA,
                       const int* __restrict__ row_start, int* __restrict__ csr) {
  int n = blockIdx.x * 256 + threadIdx.x;
  if (n >= NN) return;
  int w = row_start[n];                      // deterministic per-dst scan
  for (int e = 0; e < NEN; ++e)
    if (dstA[e] == n) csr[w++] = srcA[e];
}

// ---- gated TCN (WMMA): gg = tanh(xa@Wa0+xb@Wa1+ba)*sigmoid(xa@Wb0+xb@Wb1+bb)
__global__ __launch_bounds__(256) void k_gated(
    const float* __restrict__ x, int Tcur, int Tg, int dil,
    const float* __restrict__ Wa, const float* __restrict__ ba,
    const float* __restrict__ Wb, const float* __restrict__ bb,
    float* __restrict__ gg, int nTiles) {
  int warp = threadIdx.x >> 5, lane = threadIdx.x & 31;
  int tile = blockIdx.x * 8 + warp;
  if (tile >= nTiles) return;
  int mrow = tile * 16 + (lane & 15);
  int b = mrow / (NN * Tg), r1 = mrow % (NN * Tg);
  int n = r1 / Tg, t = r1 % Tg;
  const float* xa = x + (((long)(b * NN + n)) * Tcur + t) * 32;
  const float* xb = xa + (long)dil * 32;
  int khalf = (lane >> 4) * 2;
  int col = lane & 15;
  v8f a0 = {}, a1 = {}, b0 = {}, b1 = {};
  for (int kk = 0; kk < 8; ++kk) {
    int k0 = kk * 4, kr = k0 + khalf;
    v2f Aa = {xa[kr], xa[kr + 1]};
    v2f Ab = {xb[kr], xb[kr + 1]};
    v2f w;
    w = (v2f){Wa[(col * 32 + kr) * 2 + 0], Wa[(col * 32 + kr + 1) * 2 + 0]};
    a0 = wmma4(Aa, w, a0);
    w = (v2f){Wa[(col * 32 + kr) * 2 + 1], Wa[(col * 32 + kr + 1) * 2 + 1]};
    a0 = wmma4(Ab, w, a0);
    w = (v2f){Wa[((col + 16) * 32 + kr) * 2 + 0], Wa[((col + 16) * 32 + kr + 1) * 2 + 0]};
    a1 = wmma4(Aa, w, a1);
    w = (v2f){Wa[((col + 16) * 32 + kr) * 2 + 1], Wa[((col + 16) * 32 + kr + 1) * 2 + 1]};
    a1 = wmma4(Ab, w, a1);
    w = (v2f){Wb[(col * 32 + kr) * 2 + 0], Wb[(col * 32 + kr + 1) * 2 + 0]};
    b0 = wmma4(Aa, w, b0);
    w = (v2f){Wb[(col * 32 + kr) * 2 + 1], Wb[(col * 32 + kr + 1) * 2 + 1]};
    b0 = wmma4(Ab, w, b0);
    w = (v2f){Wb[((col + 16) * 32 + kr) * 2 + 0], Wb[((col + 16) * 32 + kr + 1) * 2 + 0]};
    b1 = wmma4(Aa, w, b1);
    w = (v2f){Wb[((col + 16) * 32 + kr) * 2 + 1], Wb[((col + 16) * 32 + kr + 1) * 2 + 1]};
    b1 = wmma4(Ab, w, b1);
  }
  int rowoff = (lane >> 4) * 8;
  for (int r = 0; r < 8; ++r) {
    long pix = (long)tile * 16 + r + rowoff;     // pixel-major == gg row index
    float ga0 = tanhf(a0[r] + ba[col]);
    float ga1 = tanhf(a1[r] + ba[col + 16]);
    float gb0 = 1.f / (1.f + __expf(-(b0[r] + bb[col])));
    float gb1 = 1.f / (1.f + __expf(-(b1[r] + bb[col + 16])));
    gg[pix * 32 + col] = ga0 * gb0;
    gg[pix * 32 + col + 16] = ga1 * gb1;
  }
}

// ----------- generic fp32 WMMA GEMM: C(M,Nout) = A(M,K) * W (+bias,relu,acc)
__global__ __launch_bounds__(256) void k_gemm(
    const float* __restrict__ A, long strideA, long offA,
    const float* __restrict__ W, int bt, const float* __restrict__ bias,
    float* __restrict__ C, long strideC, int M, int K, int Nout, int flags) {
  int warp = threadIdx.x >> 5, lane = threadIdx.x & 31;
  int ntiles = Nout >> 4;
  int tile = blockIdx.x * 8 + warp;
  int total = (M >> 4) * ntiles;
  if (tile >= total) return;
  int mt = tile / ntiles, nt = tile % ntiles;
  int mrow = mt * 16 + (lane & 15);
  int col = nt * 16 + (lane & 15);
  int khalf = (lane >> 4) * 2;
  const float* Ar = A + (long)mrow * strideA + offA;
  v8f acc = {};
  for (int k0 = 0; k0 < K; k0 += 4) {
    int kr = k0 + khalf;
    v2f a = {Ar[kr], Ar[kr + 1]};
    v2f b;
    if (bt) b = (v2f){W[(long)col * K + kr], W[(long)col * K + kr + 1]};
    else    b = (v2f){W[(long)kr * Nout + col], W[(long)(kr + 1) * Nout + col]};
    acc = wmma4(a, b, acc);
  }
  int rowoff = (lane >> 4) * 8;
  float bv = bias ? bias[col] : 0.f;
  for (int r = 0; r < 8; ++r) {
    long row = (long)mt * 16 + r + rowoff;
    float v = acc[r] + bv;
    if (flags & 1) v = fmaxf(v, 0.f);
    float* dst = C + row * strideC + col;
    if (flags & 2) v += *dst;
    *dst = v;
  }
}

// ------------- GAT attention logits: als/ald per (pixel, head) --------------
__global__ void k_attn_logits(const float* __restrict__ h,
                              const float* __restrict__ a_s,
                              const float* __restrict__ a_d,
                              float* __restrict__ als, float* __restrict__ ald,
                              int M) {
  int id = blockIdx.x * 256 + threadIdx.x;
  if (id >= M * 4) return;
  int p = id >> 2, hh = id & 3;
  const float* hp = h + (long)p * 128 + hh * 32;
  float s = 0.f, d = 0.f;
  for (int r = 0; r < 32; ++r) {
    float v = hp[r];
    s += v * a_s[hh * 32 + r];
    d += v * a_d[hh * 32 + r];
  }
  als[id] = s;
  ald[id] = d;
}

// ------ GAT softmax-aggregate (one wave per (b,t,dst)), + bias + residual ---
__global__ __launch_bounds__(256) void k_gat_agg(
    const float* __restrict__ h, const float* __restrict__ als,
    const float* __restrict__ ald, const int* __restrict__ row_start,
    const int* __restrict__ csr, const float* __restrict__ gat_b,
    const float* __restrict__ xres, int Tcur, int dil,
    float* __restrict__ xout, int Tg) {
  int warp = threadIdx.x >> 5, lane = threadIdx.x & 31;
  int idx = blockIdx.x * 8 + warp;
  int total = NB * Tg * NN;
  if (idx >= total) return;
  int n = idx % NN, bt = idx / NN;
  int t = bt % Tg, b = bt / Tg;
  long pixD = ((long)b * NN + n) * Tg + t;
  int e0 = row_start[n], e1 = row_start[n + 1];
  float out = 0.f;
  for (int hh = 0; hh < 4; ++hh) {
    float aldv = ald[pixD * 4 + hh];
    float mx = -3.0e38f;
    for (int e = e0 + lane; e < e1; e += 32) {
      int s = csr[e];
      float sc = als[(((long)b * NN + s) * Tg + t) * 4 + hh] + aldv;
      sc = sc > 0.f ? sc : 0.2f * sc;
      mx = fmaxf(mx, sc);
    }
    for (int o = 16; o > 0; o >>= 1) mx = fmaxf(mx, __shfl_xor(mx, o));
    float den = 0.f;
    for (int e = e0 + lane; e < e1; e += 32) {
      int s = csr[e];
      float sc = als[(((long)b * NN + s) * Tg + t) * 4 + hh] + aldv;
      sc = sc > 0.f ? sc : 0.2f * sc;
      den += __expf(sc - mx);
    }
    for (int o = 16; o > 0; o >>= 1) den += __shfl_xor(den, o);
    float inv = 1.f / den;
    for (int e = e0; e < e1; ++e) {     // lane = output channel, coalesced h
      int s = csr[e];
      float sc = als[(((long)b * NN + s) * Tg + t) * 4 + hh] + aldv;
      sc = sc > 0.f ? sc : 0.2f * sc;
      float alpha = __expf(sc - mx) * inv;
      out += alpha * h[(((long)b * NN + s) * Tg + t) * 128 + hh * 32 + lane];
    }
  }
  out *= 0.25f;   // mean over H=4 heads
  float res = xres[(((long)b * NN + n) * Tcur + (t + dil)) * 32 + lane];
  xout[pixD * 32 + lane] = out + gat_b[lane] + res;
}

// -------------------------- BatchNorm (per 32 channels) ---------------------
__global__ __launch_bounds__(256) void k_bn_reduce(const float* __restrict__ x,
                                                   long count,
                                                   float* __restrict__ sums) {
  __shared__ float ls[32], lss[32];
  if (threadIdx.x < 32) { ls[threadIdx.x] = 0.f; lss[threadIdx.x] = 0.f; }
  __syncthreads();
  long id = (long)blockIdx.x * 256 + threadIdx.x;
  int chan = threadIdx.x & 31;
  float v = (id < count * 32) ? x[id] : 0.f;
  atomicAdd(&ls[chan], v);
  atomicAdd(&lss[chan], v * v);
  __syncthreads();
  if (threadIdx.x < 32) {
    atomicAdd(&sums[threadIdx.x], ls[threadIdx.x]);
    atomicAdd(&sums[32 + threadIdx.x], lss[threadIdx.x]);
  }
}
__global__ void k_bn_norm(float* __restrict__ x, const float* __restrict__ sums,
                          const float* __restrict__ g,
                          const float* __restrict__ bb, long count) {
  long id = (long)blockIdx.x * 256 + threadIdx.x;
  if (id >= count * 32) return;
  int chan = (int)(id & 31);
  float cnt = (float)count;
  float mu = sums[chan] / cnt;
  float var = sums[32 + chan] / cnt - mu * mu;
  x[id] = g[chan] * (x[id] - mu) * rsqrtf(var + 1e-5f) + bb[chan];
}

// --------------------------------- tail -------------------------------------
__global__ void k_relu(float* __restrict__ p, long n) {
  long id = (long)blockIdx.x * 256 + threadIdx.x;
  if (id < n) p[id] = fmaxf(p[id], 0.f);
}
__global__ void k_end2(const float* __restrict__ y2, const float* __restrict__ w,
                       const float* __restrict__ bias, float* __restrict__ out) {
  int id = blockIdx.x * 256 + threadIdx.x;
  if (id >= NB * 12 * NN) return;
  int n = id % NN, q = id / NN;
  int o = q % 12, b = q / 12;
  const float* yp = y2 + ((long)b * NN + n) * 512;
  float acc = bias[o];
  for (int e = 0; e < 512; ++e) acc += yp[e] * w[o * 512 + e];
  out[((long)(b * 12 + o)) * NN + n] = acc;     // (B, OUT_T, N, 1)
}

extern "C" void kernel_launch(void* const* d_in, const int* in_sizes, int n_in,
                              void* d_out, int out_size, void* d_ws,
                              size_t ws_size, hipStream_t stream) {
  (void)in_sizes; (void)n_in; (void)out_size; (void)ws_size;
  const float* dynamic = (const float*)d_in[0];
  const float* stat    = (const float*)d_in[1];
  const int*   ei      = (const int*)d_in[2];
  const float* w_in    = (const float*)d_in[3];
  const float* b_in    = (const float*)d_in[4];
  const float* w_s1    = (const float*)d_in[5];
  const float* b_s1    = (const float*)d_in[6];
  const float* w_s2    = (const float*)d_in[7];
  const float* b_s2    = (const float*)d_in[8];
  const float* tcn_a_w = (const float*)d_in[9];
  const float* tcn_a_b = (const float*)d_in[10];
  const float* tcn_b_w = (const float*)d_in[11];
  const float* tcn_b_b = (const float*)d_in[12];
  const float* skip_w  = (const float*)d_in[13];
  const float* skip_b  = (const float*)d_in[14];
  const float* bn_g    = (const float*)d_in[15];
  const float* bn_b    = (const float*)d_in[16];
  const float* gat_w   = (const float*)d_in[17];
  const float* gat_as  = (const float*)d_in[18];
  const float* gat_ad  = (const float*)d_in[19];
  const float* gat_b   = (const float*)d_in[20];
  const float* end1_w  = (const float*)d_in[21];
  const float* end1_b  = (const float*)d_in[22];
  const float* end2_w  = (const float*)d_in[23];
  const float* end2_b  = (const float*)d_in[24];
  float* out = (float*)d_out;

  char* ws = (char*)d_ws;
  size_t off = 0;
  auto alloc = [&](size_t bytes) -> char* {
    char* p = ws + off;
    off = (off + bytes + 255) & ~(size_t)255;
    return p;
  };
  float* s_emb = (float*)alloc((size_t)NN * 32 * 4);
  float* x0    = (float*)alloc((size_t)NB * NN * NT * 32 * 4);
  float* x1    = (float*)alloc((size_t)NB * NN * NT * 32 * 4);
  float* gg    = (float*)alloc((size_t)NB * NN * 12 * 32 * 4);
  float* hbuf  = (float*)alloc((size_t)NB * NN * 12 * 128 * 4);  // reused as y2
  float* als   = (float*)alloc((size_t)NB * NN * 12 * 4 * 4);
  float* ald   = (float*)alloc((size_t)NB * NN * 12 * 4 * 4);
  float* skip  = (float*)alloc((size_t)NB * NN * 256 * 4);
  float* bns   = (float*)alloc(64 * 4);
  int* srcA = (int*)alloc((size_t)NEN * 4);
  int* dstA = (int*)alloc((size_t)NEN * 4);
  int* deg  = (int*)alloc((size_t)NN * 4);
  int* rowS = (int*)alloc((size_t)(NN + 1) * 4);
  int* csr  = (int*)alloc((size_t)NEN * 4);

  // CSR build (deterministic)
  hipMemsetAsync(deg, 0, NN * 4, stream);
  k_edges<<<(NEN + 255) / 256, 256, 0, stream>>>(ei, srcA, dstA);
  k_deg<<<(NEN + 255) / 256, 256, 0, stream>>>(dstA, deg);
  k_scan<<<1, 1024, 0, stream>>>(deg, rowS);
  k_fill<<<(NN + 255) / 256, 256, 0, stream>>>(srcA, dstA, rowS, csr);

  // input path
  k_static<<<NN, 64, 0, stream>>>(stat, w_s1, b_s1, w_s2, b_s2, s_emb);
  {
    long tot = (long)NB * NN * NT * 32;
    k_input<<<(int)((tot + 255) / 256), 256, 0, stream>>>(dynamic, w_in, b_in,
                                                          s_emb, x0);
  }

  float* xin = x0; float* xout = x1;
  int Tcur = NT;
  const int dils[8] = {1, 2, 1, 2, 1, 2, 1, 2};
  for (int k = 0; k < 8; ++k) {
    int d = dils[k], Tg = Tcur - d;
    int M = NB * NN * Tg;
    int nT = M / 16;
    k_gated<<<(nT + 7) / 8, 256, 0, stream>>>(
        xin, Tcur, Tg, d, tcn_a_w + (size_t)k * 32 * 32 * 2, tcn_a_b + k * 32,
        tcn_b_w + (size_t)k * 32 * 32 * 2, tcn_b_b + k * 32, gg, nT);
    {   // skip projection only at t = Tg-1 (only tap the output reads)
      int tiles = (NB * NN / 16) * (256 / 16);
      k_gemm<<<(tiles + 7) / 8, 256, 0, stream>>>(
          gg, (long)Tg * 32, (long)(Tg - 1) * 32, skip_w + (size_t)k * 256 * 32,
          1, skip_b + k * 256, skip, 256, NB * NN, 32, 256, (k == 0) ? 0 : 2);
    }
    if (k < 7) {  // layer 8's GAT/BN output is dead — skip it
      int tiles = (M / 16) * (128 / 16);
      k_gemm<<<(tiles + 7) / 8, 256, 0, stream>>>(
          gg, 32, 0, gat_w + (size_t)k * 32 * 128, 0, nullptr, hbuf, 128, M, 32,
          128, 0);
      k_attn_logits<<<(M * 4 + 255) / 256, 256, 0, stream>>>(
          hbuf, gat_as + k * 128, gat_ad + k * 128, als, ald, M);
      k_gat_agg<<<(M + 7) / 8, 256, 0, stream>>>(
          hbuf, als, ald, rowS, csr, gat_b + k * 32, xin, Tcur, d, xout, Tg);
      hipMemsetAsync(bns, 0, 64 * 4, stream);
      long elems = (long)M * 32;
      k_bn_reduce<<<(int)((elems + 255) / 256), 256, 0, stream>>>(xout, M, bns);
      k_bn_norm<<<(int)((elems + 255) / 256), 256, 0, stream>>>(
          xout, bns, bn_g + k * 32, bn_b + k * 32, M);
      float* tmp = xin; xin = xout; xout = tmp;
    }
    Tcur = Tg;
  }

  // tail: relu(skip) -> end1 (WMMA, relu) -> end2 -> output transpose
  k_relu<<<(NB * NN * 256 + 255) / 256, 256, 0, stream>>>(skip,
                                                          (long)NB * NN * 256);
  {
    int tiles = (NB * NN / 16) * (512 / 16);
    k_gemm<<<(tiles + 7) / 8, 256, 0, stream>>>(skip, 256, 0, end1_w, 1, end1_b,
                                                hbuf, 512, NB * NN, 256, 512, 1);
  }
  k_end2<<<(NB * 12 * NN + 255) / 256, 256, 0, stream>>>(hbuf, end2_w, end2_b,
                                                         out);
}